// RWLoss_50534585204955
// MI455X (gfx1250) — compile-verified
//
#include <hip/hip_runtime.h>
#include <hip/hip_bf16.h>

// ---------------------------------------------------------------------------
// RWLoss for MI455X (gfx1250, wave32).
// Pipeline:
//   K0: zero per-image max accumulators (uint-bit float max)
//   K1: per-column two-direction capped scan -> g^2            (VALU)
//   K2: per-row exact min-plus + sqrt + per-image max          (VALU + LDS)
//   K3: softmax * rrwmap, wave-sum via V_WMMA_F32_16X16X4_F32  (WMMA)
//   K4: deterministic final reduction -> scalar mean
// ---------------------------------------------------------------------------

#define RW_B 8
#define RW_C 4
#define RW_H 256
#define RW_W 256
#define RW_N (RW_B * RW_C)            // 32 images
#define RW_HW (RW_H * RW_W)           // 65536
#define RW_NHW (RW_N * RW_HW)         // 2,097,152 floats in g2/d buffer
#define RW_INF 1.0e6f

// K3 geometry: B*H*W = 524288 positions, 2 per thread -> 262144 threads
#define K3_THREADS 256
#define K3_BLOCKS 1024                // 1024*256 = 262144
#define K3_STRIDE (K3_BLOCKS * K3_THREADS)
#define N_PARTIALS (K3_BLOCKS * (K3_THREADS / 32))   // 8192 wave partials

typedef __attribute__((ext_vector_type(2))) float v2f;
typedef __attribute__((ext_vector_type(8))) float v8f;

// ---------------------------------------------------------------- K0: init
__global__ void rw_init_kernel(unsigned int* mx) {
    if (threadIdx.x < RW_N) mx[threadIdx.x] = 0u;   // bits of +0.0f
}

// ----------------------------------------------- K1: column distance -> g^2
// grid: RW_N blocks of RW_W threads. block = image n, thread = column w.
__global__ void rw_coldist_kernel(const int* __restrict__ y, float* __restrict__ g2) {
    const int n = blockIdx.x;          // image index in [0, 32)
    const int w = threadIdx.x;         // column
    const int b = n / RW_C;
    const int c = n % RW_C;
    const int* yb = y + b * RW_HW;
    float* gp = g2 + n * RW_HW;

    // Downward scan: d = (onehot==0) ? 0 : min(carry+1, INF)
    float carry = RW_INF;
    for (int h = 0; h < RW_H; ++h) {
        const int lab = yb[h * RW_W + w];
        const float dv = (lab == c) ? fminf(carry + 1.0f, RW_INF) : 0.0f;
        gp[h * RW_W + w] = dv;
        carry = dv;
    }
    // Upward scan, combine, square. Same-wave store->load same address is ordered.
    carry = RW_INF;
    for (int h = RW_H - 1; h >= 0; --h) {
        const int lab = yb[h * RW_W + w];
        const float dv = (lab == c) ? fminf(carry + 1.0f, RW_INF) : 0.0f;
        const float g = fminf(gp[h * RW_W + w], dv);
        gp[h * RW_W + w] = g * g;
        carry = dv;
    }
}

// -------------------------------- K2: exact per-row min-plus (in place) + max
// grid: RW_N*RW_H blocks of RW_W threads. block = one row of one image.
__global__ void rw_minplus_kernel(float* __restrict__ g2d, unsigned int* __restrict__ mx) {
    __shared__ float srow[RW_W];
    __shared__ float sred[RW_W];

    const int row = blockIdx.x;                  // n*H + i
    const int n   = row / RW_H;
    const int x   = threadIdx.x;
    float* rp = g2d + (size_t)row * RW_W;

    srow[x] = rp[x];
    __syncthreads();

    // D2[x] = min_j (g2[j] + (x-j)^2)  -- brute force, matches reference exactly
    float best = srow[x];                        // j == x term
    #pragma unroll 8
    for (int j = 0; j < RW_W; ++j) {
        const float dx = (float)(x - j);
        best = fminf(best, fmaf(dx, dx, srow[j]));
    }
    const float dv = sqrtf(best);
    rp[x] = dv;                                  // overwrite g2 row with d

    // Block max-reduce, then one exact uint-bit atomic max per block (d >= 0)
    sred[x] = dv;
    __syncthreads();
    for (int st = RW_W / 2; st > 0; st >>= 1) {
        if (x < st) sred[x] = fmaxf(sred[x], sred[x + st]);
        __syncthreads();
    }
    if (x == 0) atomicMax(mx + n, __float_as_uint(sred[0]));
}

// --------------------- K3: softmax * rrwmap + WMMA wave reduction -> partials
// grid: K3_BLOCKS x K3_THREADS, no divergence before the WMMA (EXEC all ones).
__global__ void rw_fuse_kernel(const float* __restrict__ x,
                               const float* __restrict__ d,
                               const unsigned int* __restrict__ mxbits,
                               float* __restrict__ partials) {
    const int g = blockIdx.x * K3_THREADS + threadIdx.x;   // [0, 262144)

    float acc[2];
    #pragma unroll
    for (int s = 0; s < 2; ++s) {
        const int pos = g + s * K3_STRIDE;                 // [0, B*H*W)
        const int b   = pos >> 16;                         // / (H*W)
        const int hw  = pos & (RW_HW - 1);

        // softmax over the 4 channels at this pixel
        float xv[RW_C];
        #pragma unroll
        for (int c = 0; c < RW_C; ++c)
            xv[c] = x[((b * RW_C + c) << 16) + hw];
        float m = xv[0];
        #pragma unroll
        for (int c = 1; c < RW_C; ++c) m = fmaxf(m, xv[c]);
        float e[RW_C], es = 0.0f;
        #pragma unroll
        for (int c = 0; c < RW_C; ++c) { e[c] = __expf(xv[c] - m); es += e[c]; }
        const float inv = 1.0f / es;

        float a = 0.0f;
        #pragma unroll
        for (int c = 0; c < RW_C; ++c) {
            const int nidx = b * RW_C + c;
            const float dv = d[(nidx << 16) + hw];
            const float mxv = __uint_as_float(mxbits[nidx]);
            float r = -dv / (mxv + 1e-15f);
            r = (r == 0.0f) ? 1.0f : r;                    // -0.0 == 0 -> 1
            a = fmaf(e[c] * inv, r, a);
        }
        acc[s] = a;
    }

    // Wave-wide sum of 64 values via one FP32 WMMA: D = A * ones + 0.
    // A (16x4) holds the wave's 64 partials (2 VGPRs x 32 lanes); with B = 1s,
    // every column of D is the 16 row-sums; column total = grand total.
    v2f av; av.x = acc[0]; av.y = acc[1];
    v2f ones; ones.x = 1.0f; ones.y = 1.0f;
    v8f cz = {};
    v8f ds = __builtin_amdgcn_wmma_f32_16x16x4_f32(
        /*neg_a=*/false, av, /*neg_b=*/false, ones,
        /*c_mod=*/(short)0, cz, /*reuse_a=*/false, /*reuse_b=*/false);

    float loc = ds[0] + ds[1] + ds[2] + ds[3] + ds[4] + ds[5] + ds[6] + ds[7];
    // lane L holds half of column (L%16); pair with lane L^16 for the full sum
    const float tot = loc + __shfl_xor(loc, 16, 32);

    const int lane = threadIdx.x & 31;
    const int wave = (blockIdx.x * K3_THREADS + threadIdx.x) >> 5;
    if (lane == 0) partials[wave] = tot;
}

// ------------------------------------- K4: deterministic final sum -> d_out
__global__ void rw_final_kernel(const float* __restrict__ partials, float* __restrict__ out) {
    __shared__ float red[256];
    const int t = threadIdx.x;
    float s = 0.0f;
    for (int i = t; i < N_PARTIALS; i += 256) s += partials[i];  // fixed order
    red[t] = s;
    __syncthreads();
    for (int st = 128; st > 0; st >>= 1) {
        if (t < st) red[t] += red[t + st];
        __syncthreads();
    }
    if (t == 0) out[0] = red[0] * (1.0f / (float)(RW_B * RW_C * RW_H * RW_W));
}

// ---------------------------------------------------------------------------
extern "C" void kernel_launch(void* const* d_in, const int* in_sizes, int n_in,
                              void* d_out, int out_size, void* d_ws, size_t ws_size,
                              hipStream_t stream) {
    const float* x = (const float*)d_in[0];     // [B, C, H, W] f32
    const int*   y = (const int*)d_in[1];       // [B, 1, H, W] i32
    float* out = (float*)d_out;                 // scalar

    char* ws = (char*)d_ws;
    float*        g2d      = (float*)ws;                                  // 8 MiB
    unsigned int* mx       = (unsigned int*)(ws + (size_t)RW_NHW * 4);    // 32 u32
    float*        partials = (float*)(ws + (size_t)RW_NHW * 4 + 256);     // 8192 f32

    rw_init_kernel<<<1, 64, 0, stream>>>(mx);
    rw_coldist_kernel<<<RW_N, RW_W, 0, stream>>>(y, g2d);
    rw_minplus_kernel<<<RW_N * RW_H, RW_W, 0, stream>>>(g2d, mx);
    rw_fuse_kernel<<<K3_BLOCKS, K3_THREADS, 0, stream>>>(x, g2d, mx, partials);
    rw_final_kernel<<<1, 256, 0, stream>>>(partials, out);
}